// Het_En_49323404427449
// MI455X (gfx1250) — compile-verified
//
#include <hip/hip_runtime.h>
#include <math.h>

#define D 128
#define NA 40000
#define NPN 60000
#define NV 10000
#define NE 400000

typedef __attribute__((ext_vector_type(2))) float v2f;
typedef __attribute__((ext_vector_type(8))) float v8f;

// ---------------------------------------------------------------------------
// Generic GEMM: out[M,128] = relu( sum_p X_p[M,128] @ W[p*128:(p+1)*128, 0:128] + bias )
// One 16x16 output tile per wave via V_WMMA_F32_16X16X4_F32.
// Block = 256 threads = 8 waves -> 16 rows x 128 cols per block. M % 16 == 0.
// NPARTS templated so part pointers resolve at compile time (keeps loads in
// the global address space -> global_load_b64, no flat/scratch traffic).
// ---------------------------------------------------------------------------
template <int NPARTS>
__global__ void __launch_bounds__(256)
wmma_gemm_bias_relu(const float* __restrict__ x0, const float* __restrict__ x1,
                    const float* __restrict__ x2, const float* __restrict__ x3,
                    const float* __restrict__ x4, const float* __restrict__ x5,
                    const float* __restrict__ W,     // [NPARTS*128, 128] row-major
                    const float* __restrict__ bias,  // [128]
                    float* __restrict__ out, int M)
{
    const int wave = threadIdx.x >> 5;       // 0..7 -> column tile
    const int lane = threadIdx.x & 31;
    const int half = lane >> 4;              // 0: lanes 0-15, 1: lanes 16-31
    const int l16  = lane & 15;
    const int row0 = blockIdx.x * 16;
    const int col0 = wave * 16;
    if (row0 >= M) return;

    v8f acc = {};
#pragma unroll
    for (int p = 0; p < NPARTS; ++p) {
        const float* __restrict__ X =
            (p == 0) ? x0 : (p == 1) ? x1 : (p == 2) ? x2 :
            (p == 3) ? x3 : (p == 4) ? x4 : x5;
        const float* __restrict__ Wp   = W + (size_t)p * D * D;
        const float* __restrict__ arow = X + (size_t)(row0 + l16) * D + half * 2;
#pragma unroll
        for (int k = 0; k < D; k += 4) {
            // A fragment (16x4 f32): v[0]=A[row][k+2*half], v[1]=A[row][k+2*half+1]
            v2f a;
            a.x = arow[k + 0];
            a.y = arow[k + 1];
            // B fragment (4x16 f32): v[0]=W[k+2*half][col], v[1]=W[k+2*half+1][col]
            const float* __restrict__ bp = Wp + (size_t)(k + half * 2) * D + col0 + l16;
            v2f b;
            b.x = bp[0];
            b.y = bp[D];
            acc = __builtin_amdgcn_wmma_f32_16x16x4_f32(
                false, a, false, b, (short)0, acc, false, false);
        }
    }

    // Epilogue: C/D layout -> row = row0 + half*8 + r, col = col0 + l16
    const float bc = bias[col0 + l16];
    float* __restrict__ orow = out + (size_t)(row0 + half * 8) * D + col0 + l16;
#pragma unroll
    for (int r = 0; r < 8; ++r) {
        float v = acc[r] + bc;
        v = v > 0.0f ? v : 0.0f;
        orow[(size_t)r * D] = v;
    }
}

// ---------------------------------------------------------------------------
// In-place row L2 normalize: x / max(||x||, 1e-12). One wave per 128-wide row.
// ---------------------------------------------------------------------------
__global__ void __launch_bounds__(256)
l2norm_rows(float* __restrict__ buf, int M)
{
    const int row  = blockIdx.x * (blockDim.x >> 5) + (threadIdx.x >> 5);
    const int lane = threadIdx.x & 31;
    if (row >= M) return;
    float4* p = (float4*)(buf + (size_t)row * D);
    float4 v = p[lane];
    float ss = v.x * v.x + v.y * v.y + v.z * v.z + v.w * v.w;
#pragma unroll
    for (int off = 16; off > 0; off >>= 1) ss += __shfl_xor(ss, off, 32);
    const float inv = 1.0f / fmaxf(sqrtf(ss), 1e-12f);
    v.x *= inv; v.y *= inv; v.z *= inv; v.w *= inv;
    p[lane] = v;
}

// ---------------------------------------------------------------------------
// Scatter-mean accumulation: aggr[src] += xt[dst]; cnt[src] += 1.
// One wave per edge, 4 columns per lane.  edge[0][e]=src, edge[1][e]=dst.
// ---------------------------------------------------------------------------
__global__ void __launch_bounds__(256)
scatter_accum(const int* __restrict__ edge, int E,
              const float* __restrict__ xt,
              float* __restrict__ aggr, float* __restrict__ cnt)
{
    const long long gid = (long long)blockIdx.x * blockDim.x + threadIdx.x;
    const long long e = gid >> 5;
    const int lane = (int)(gid & 31);
    if (e >= E) return;
    const int src = edge[e];
    const int dst = edge[(size_t)E + e];
    const float4 v = ((const float4*)(xt + (size_t)dst * D))[lane];
    float* __restrict__ ap = aggr + (size_t)src * D + lane * 4;
    atomicAdd(ap + 0, v.x);
    atomicAdd(ap + 1, v.y);
    atomicAdd(ap + 2, v.z);
    atomicAdd(ap + 3, v.w);
    if (lane == 0) atomicAdd(cnt + src, 1.0f);
}

// ---------------------------------------------------------------------------
// Semantic attention fusion. One wave per row.
//   h_X = aggr_X/max(cnt_X,1);  s_X = exp(lrelu(h_X.u_lo + x_node.u_hi))
//   comb = (sa*ha + sb*hb + sc*hc) / (sa+sb+sc)
// ---------------------------------------------------------------------------
__device__ __forceinline__ float wave_sum(float v) {
#pragma unroll
    for (int off = 16; off > 0; off >>= 1) v += __shfl_xor(v, off, 32);
    return v;
}

__global__ void __launch_bounds__(256)
attention_comb(const float* __restrict__ x_node,
               const float* __restrict__ aggrA, const float* __restrict__ cntA,
               const float* __restrict__ aggrP, const float* __restrict__ cntP,
               const float* __restrict__ aggrV, const float* __restrict__ cntV,
               const float* __restrict__ u,     // [256]
               float* __restrict__ comb, int M)
{
    const int row  = blockIdx.x * (blockDim.x >> 5) + (threadIdx.x >> 5);
    const int lane = threadIdx.x & 31;
    if (row >= M) return;
    const size_t base = (size_t)row * D;

    float4 xn  = ((const float4*)(x_node + base))[lane];
    float4 ulo = ((const float4*)u)[lane];
    float4 uhi = ((const float4*)(u + D))[lane];

    const float ia = 1.0f / fmaxf(cntA[row], 1.0f);
    const float ib = 1.0f / fmaxf(cntP[row], 1.0f);
    const float ic = 1.0f / fmaxf(cntV[row], 1.0f);

    float4 ha = ((const float4*)(aggrA + base))[lane];
    float4 hb = ((const float4*)(aggrP + base))[lane];
    float4 hc = ((const float4*)(aggrV + base))[lane];
    ha.x *= ia; ha.y *= ia; ha.z *= ia; ha.w *= ia;
    hb.x *= ib; hb.y *= ib; hb.z *= ib; hb.w *= ib;
    hc.x *= ic; hc.y *= ic; hc.z *= ic; hc.w *= ic;

    const float px = xn.x * uhi.x + xn.y * uhi.y + xn.z * uhi.z + xn.w * uhi.w;
    float da = wave_sum(ha.x * ulo.x + ha.y * ulo.y + ha.z * ulo.z + ha.w * ulo.w + px);
    float db = wave_sum(hb.x * ulo.x + hb.y * ulo.y + hb.z * ulo.z + hb.w * ulo.w + px);
    float dc = wave_sum(hc.x * ulo.x + hc.y * ulo.y + hc.z * ulo.z + hc.w * ulo.w + px);

    const float sa = expf(da > 0.0f ? da : 0.01f * da);
    const float sb = expf(db > 0.0f ? db : 0.01f * db);
    const float sc = expf(dc > 0.0f ? dc : 0.01f * dc);
    const float inv = 1.0f / (sa + sb + sc);

    float4 o;
    o.x = (sa * ha.x + sb * hb.x + sc * hc.x) * inv;
    o.y = (sa * ha.y + sb * hb.y + sc * hc.y) * inv;
    o.z = (sa * ha.z + sb * hb.z + sc * hc.z) * inv;
    o.w = (sa * ha.w + sb * hb.w + sc * hc.w) * inv;
    ((float4*)(comb + base))[lane] = o;
}

// ---------------------------------------------------------------------------
// Host-side orchestration
// ---------------------------------------------------------------------------
static inline void launch_gemm1(const float* x0, const float* W, const float* b,
                                float* out, int M, hipStream_t s)
{
    wmma_gemm_bias_relu<1><<<M / 16, 256, 0, s>>>(
        x0, x0, x0, x0, x0, x0, W, b, out, M);
}
static inline void launch_gemm2(const float* x0, const float* x1,
                                const float* W, const float* b,
                                float* out, int M, hipStream_t s)
{
    wmma_gemm_bias_relu<2><<<M / 16, 256, 0, s>>>(
        x0, x1, x1, x1, x1, x1, W, b, out, M);
}
static inline void launch_gemm6(const float* x0, const float* x1, const float* x2,
                                const float* x3, const float* x4, const float* x5,
                                const float* W, const float* b,
                                float* out, int M, hipStream_t s)
{
    wmma_gemm_bias_relu<6><<<M / 16, 256, 0, s>>>(
        x0, x1, x2, x3, x4, x5, W, b, out, M);
}

extern "C" void kernel_launch(void* const* d_in, const int* in_sizes, int n_in,
                              void* d_out, int out_size, void* d_ws, size_t ws_size,
                              hipStream_t stream)
{
    (void)in_sizes; (void)n_in; (void)out_size; (void)ws_size;

    // Inputs (setup_inputs insertion order, params flattened depth-first)
    const float* a_net   = (const float*)d_in[0];
    const float* a_text  = (const float*)d_in[1];
    const float* p_title = (const float*)d_in[2];
    const float* p_abs   = (const float*)d_in[3];
    const float* p_net   = (const float*)d_in[4];
    const float* p_a_net = (const float*)d_in[5];
    const float* p_p_net = (const float*)d_in[6];
    const float* p_v_net = (const float*)d_in[7];
    const float* v_net   = (const float*)d_in[8];
    const float* v_text  = (const float*)d_in[9];
    const int*   e_aa    = (const int*)d_in[10];
    const int*   e_ap    = (const int*)d_in[11];
    const int*   e_av    = (const int*)d_in[12];
    // d_in[13..18]: e_pa..e_vv (dead code in reference)
    const float* aconW   = (const float*)d_in[19];
    const float* aconB   = (const float*)d_in[20];
    const float* pconW   = (const float*)d_in[21];
    const float* pconB   = (const float*)d_in[22];
    const float* vconW   = (const float*)d_in[23];
    const float* vconB   = (const float*)d_in[24];
    const float* aggAW   = (const float*)d_in[25];
    const float* aggAB   = (const float*)d_in[26];
    const float* aggPW   = (const float*)d_in[27];
    const float* aggPB   = (const float*)d_in[28];
    const float* aggVW   = (const float*)d_in[29];
    const float* aggVB   = (const float*)d_in[30];
    const float* u       = (const float*)d_in[31];
    const float* linW    = (const float*)d_in[32];
    const float* linB    = (const float*)d_in[33];
    // d_in[34..]: p_het / v_het params (dead code in reference)

    float* ws = (float*)d_ws;
    float* a_cont = ws;                         // NA*D
    float* p_cont = a_cont + (size_t)NA * D;    // NPN*D
    float* v_cont = p_cont + (size_t)NPN * D;   // NV*D
    float* xt_a   = v_cont + (size_t)NV * D;    // NA*D
    float* xt_p   = xt_a   + (size_t)NA * D;    // NPN*D
    float* xt_v   = xt_p   + (size_t)NPN * D;   // NV*D
    float* aggrA  = xt_v   + (size_t)NV * D;    // NA*D
    float* aggrP  = aggrA  + (size_t)NA * D;    // NA*D
    float* aggrV  = aggrP  + (size_t)NA * D;    // NA*D
    float* cntA   = aggrV  + (size_t)NA * D;    // NA
    float* cntP   = cntA + NA;                  // NA
    float* cntV   = cntP + NA;                  // NA
    float* comb   = cntV + NA;                  // NA*D

    float* outF = (float*)d_out;
    const int rowsBlk = 256 / 32; // waves per block for row-parallel kernels

    // 1) Content encoders: relu(concat @ W + b) then L2 norm
    launch_gemm2(a_net, a_text, aconW, aconB, a_cont, NA, stream);
    l2norm_rows<<<(NA + rowsBlk - 1) / rowsBlk, 256, 0, stream>>>(a_cont, NA);

    launch_gemm6(p_title, p_abs, p_net, p_a_net, p_p_net, p_v_net,
                 pconW, pconB, p_cont, NPN, stream);
    l2norm_rows<<<(NPN + rowsBlk - 1) / rowsBlk, 256, 0, stream>>>(p_cont, NPN);

    launch_gemm2(v_net, v_text, vconW, vconB, v_cont, NV, stream);
    l2norm_rows<<<(NV + rowsBlk - 1) / rowsBlk, 256, 0, stream>>>(v_cont, NV);

    // 2) Neighbor transforms xt = relu(x_cont @ W_agg + b)
    launch_gemm1(a_cont, aggAW, aggAB, xt_a, NA, stream);
    launch_gemm1(p_cont, aggPW, aggPB, xt_p, NPN, stream);
    launch_gemm1(v_cont, aggVW, aggVB, xt_v, NV, stream);

    // 3) Zero accumulators (aggrA..cntV contiguous), then scatter-mean
    hipMemsetAsync(aggrA, 0, ((size_t)3 * NA * D + 3 * NA) * sizeof(float), stream);
    {
        const long long threads = (long long)NE * 32;
        const int blocks = (int)((threads + 255) / 256);
        scatter_accum<<<blocks, 256, 0, stream>>>(e_aa, NE, xt_a, aggrA, cntA);
        scatter_accum<<<blocks, 256, 0, stream>>>(e_ap, NE, xt_p, aggrP, cntP);
        scatter_accum<<<blocks, 256, 0, stream>>>(e_av, NE, xt_v, aggrV, cntV);
    }

    // 4) Attention fusion -> comb
    attention_comb<<<(NA + rowsBlk - 1) / rowsBlk, 256, 0, stream>>>(
        a_cont, aggrA, cntA, aggrP, cntP, aggrV, cntV, u, comb, NA);

    // 5) Final: l2norm(relu(concat([a_cont, comb]) @ lin_W + lin_b)) -> d_out
    launch_gemm2(a_cont, comb, linW, linB, outF, NA, stream);
    l2norm_rows<<<(NA + rowsBlk - 1) / rowsBlk, 256, 0, stream>>>(outF, NA);
}